// ASAP_35622458753573
// MI455X (gfx1250) — compile-verified
//
#include <hip/hip_runtime.h>
#include <hip/hip_bf16.h>

// Problem constants (from reference)
#define GN_N    50000
#define GN_E    800000
#define GN_FIN  128
#define GN_H    64
#define GN_G    256
#define GN_C    10

typedef __attribute__((ext_vector_type(2))) float v2f;
typedef __attribute__((ext_vector_type(8))) float v8f;

// ---------------------------------------------------------------------------
// fp32 WMMA: D(16x16) = A(16x4) * B(4x16) + C, striped across a wave32.
// A layout (ISA 7.12.2): lane L holds row M=L&15; v0 = K = (L>=16?2:0), v1 = K+1
// B layout (mirror):     lane L holds col N=L&15; v0 = K = (L>=16?2:0), v1 = K+1
// C/D layout:            reg r -> M = r + (L>=16?8:0), N = L&15
// ---------------------------------------------------------------------------
static __device__ __forceinline__ v8f wmma_f32_16x16x4(v2f a, v2f b, v8f c) {
#if __has_builtin(__builtin_amdgcn_wmma_f32_16x16x4_f32)
    return __builtin_amdgcn_wmma_f32_16x16x4_f32(
        /*neg_a=*/false, a, /*neg_b=*/false, b,
        /*c_mod=*/(short)0, c, /*reuse_a=*/false, /*reuse_b=*/false);
#else
    // compile-only fallback (should not be taken on gfx1250)
    c[0] += a.x * b.x + a.y * b.y;
    return c;
#endif
}

// ---------------------------------------------------------------------------
__global__ void ASAP_zero_kernel(float* p, long long n) {
    long long i = (long long)blockIdx.x * blockDim.x + threadIdx.x;
    long long stride = (long long)gridDim.x * blockDim.x;
    for (; i < n; i += stride) p[i] = 0.0f;
}

__global__ void ASAP_count_kernel(const int* __restrict__ idx, float* __restrict__ cnt, int n) {
    int i = blockIdx.x * blockDim.x + threadIdx.x;
    if (i < n) atomicAdd(&cnt[idx[i]], 1.0f);
}

// One float4 chunk of one edge-row per thread. chunkShift: log2(F/4).
__global__ void ASAP_edge_scatter(const float* __restrict__ feat,
                                  const int* __restrict__ src,
                                  const int* __restrict__ dst,
                                  float* __restrict__ agg,
                                  long long nWork, int chunkShift, int F) {
    long long t = (long long)blockIdx.x * blockDim.x + threadIdx.x;
    if (t >= nWork) return;
    int e = (int)(t >> chunkShift);
    int c = (int)(t & ((1 << chunkShift) - 1));
    int s = src[e];
    int d = dst[e];
    const float4 v = *(const float4*)(feat + (long long)s * F + c * 4);
    float* p = agg + (long long)d * F + c * 4;
    atomicAdd(p + 0, v.x);
    atomicAdd(p + 1, v.y);
    atomicAdd(p + 2, v.z);
    atomicAdd(p + 3, v.w);
}

// out[nRows,64] = relu( (A1/deg) @ W1 + A2 @ W2 + bias ), fp32 WMMA.
// A1:[nRows,k1] scaled per-row by 1/max(deg,1); A2:[nRows,k2]; W:[k,64] row-major.
__global__ __launch_bounds__(256)
void ASAP_gemm_dual(const float* __restrict__ A1, const float* __restrict__ W1, int k1,
                    const float* __restrict__ A2, const float* __restrict__ W2, int k2,
                    const float* __restrict__ deg, const float* __restrict__ bias,
                    float* __restrict__ out, int nRows) {
    const int lane = threadIdx.x & 31;
    const int wave = (int)((blockIdx.x * blockDim.x + threadIdx.x) >> 5);
    const int numTiles = (nRows >> 4) * (GN_H / 16);   // wave-uniform
    if (wave >= numTiles) return;                       // uniform exit; EXEC stays all-1s

    const int tm = wave / (GN_H / 16);
    const int tn = wave % (GN_H / 16);
    const int rowBase = tm * 16;
    const int colBase = tn * 16;
    const int aRow = rowBase + (lane & 15);
    const int bCol = colBase + (lane & 15);
    const int kHalf = (lane >> 4) * 2;                  // 0 or 2

    const float inv = 1.0f / fmaxf(deg[aRow], 1.0f);

    v8f acc = {};

    // Part 1: mean-aggregated neighbors @ W_rel (scale fused into A fragment)
    {
        const float* ap = A1 + (long long)aRow * k1;
        for (int k0 = 0; k0 < k1; k0 += 4) {
            v2f a, b;
            a.x = ap[k0 + kHalf] * inv;
            a.y = ap[k0 + kHalf + 1] * inv;
            b.x = W1[(k0 + kHalf) * GN_H + bCol];
            b.y = W1[(k0 + kHalf + 1) * GN_H + bCol];
            acc = wmma_f32_16x16x4(a, b, acc);
        }
    }
    // Part 2: root features @ W_root
    {
        const float* ap = A2 + (long long)aRow * k2;
        for (int k0 = 0; k0 < k2; k0 += 4) {
            v2f a, b;
            a.x = ap[k0 + kHalf];
            a.y = ap[k0 + kHalf + 1];
            b.x = W2[(k0 + kHalf) * GN_H + bCol];
            b.y = W2[(k0 + kHalf + 1) * GN_H + bCol];
            acc = wmma_f32_16x16x4(a, b, acc);
        }
    }

    // Epilogue: bias + ReLU, scatter per C/D layout
    const float bv = bias[bCol];
    const int mOff = (lane >> 4) * 8;
    for (int r = 0; r < 8; ++r) {
        float v = acc[r] + bv;
        out[(long long)(rowBase + mOff + r) * GN_H + bCol] = fmaxf(v, 0.0f);
    }
}

__global__ void ASAP_pool_scatter(const float* __restrict__ h,
                                  const int* __restrict__ batch,
                                  float* __restrict__ pool, int n) {
    int t = blockIdx.x * blockDim.x + threadIdx.x;
    if (t >= n * GN_H) return;
    int i = t >> 6;          // node
    int f = t & 63;          // feature
    atomicAdd(&pool[batch[i] * GN_H + f], h[t]);
}

__global__ void ASAP_pool_finalize(float* __restrict__ pool1, float* __restrict__ pool2,
                                   const float* __restrict__ bcount) {
    int t = blockIdx.x * blockDim.x + threadIdx.x;
    if (t >= GN_G * GN_H) return;
    float inv = 1.0f / fmaxf(bcount[t >> 6], 1.0f);
    pool1[t] *= inv;
    pool2[t] *= inv;
}

// One block (64 threads) per graph: z = relu([p1|p2]@Wl1 + bl1); logits = z@Wl2 + bl2; log_softmax
__global__ void ASAP_head(const float* __restrict__ pool1, const float* __restrict__ pool2,
                          const float* __restrict__ Wl1, const float* __restrict__ bl1,
                          const float* __restrict__ Wl2, const float* __restrict__ bl2,
                          float* __restrict__ out) {
    __shared__ float z[GN_H];
    __shared__ float logit[16];
    const int g = blockIdx.x;
    const int t = threadIdx.x;   // 0..63

    const float* p1 = pool1 + g * GN_H;
    const float* p2 = pool2 + g * GN_H;
    float acc = bl1[t];
    for (int k = 0; k < GN_H; ++k) acc += p1[k] * Wl1[k * GN_H + t];
    for (int k = 0; k < GN_H; ++k) acc += p2[k] * Wl1[(GN_H + k) * GN_H + t];
    z[t] = fmaxf(acc, 0.0f);
    __syncthreads();

    if (t < GN_C) {
        float l = bl2[t];
        for (int k = 0; k < GN_H; ++k) l += z[k] * Wl2[k * GN_C + t];
        logit[t] = l;
    }
    __syncthreads();

    if (t < GN_C) {
        float m = -1e30f;
        for (int c = 0; c < GN_C; ++c) m = fmaxf(m, logit[c]);
        float s = 0.0f;
        for (int c = 0; c < GN_C; ++c) s += __expf(logit[c] - m);
        out[g * GN_C + t] = logit[t] - m - __logf(s);
    }
}

// ---------------------------------------------------------------------------
extern "C" void kernel_launch(void* const* d_in, const int* in_sizes, int n_in,
                              void* d_out, int out_size, void* d_ws, size_t ws_size,
                              hipStream_t stream) {
    const float* x       = (const float*)d_in[0];
    const int*   ei      = (const int*)d_in[1];
    const int*   batch   = (const int*)d_in[2];
    const float* W1_rel  = (const float*)d_in[3];
    const float* b1_rel  = (const float*)d_in[4];
    const float* W1_root = (const float*)d_in[5];
    const float* W2_rel  = (const float*)d_in[6];
    const float* b2_rel  = (const float*)d_in[7];
    const float* W2_root = (const float*)d_in[8];
    const float* Wl1     = (const float*)d_in[9];
    const float* bl1     = (const float*)d_in[10];
    const float* Wl2     = (const float*)d_in[11];
    const float* bl2     = (const float*)d_in[12];

    const int* src = ei;            // edge_index[0, :]
    const int* dst = ei + GN_E;     // edge_index[1, :]

    // Workspace layout (floats). Buffers needing zeroing are contiguous first.
    float* ws    = (float*)d_ws;
    float* deg   = ws;                                    // N
    float* bc    = deg   + GN_N;                          // G
    float* agg1  = bc    + GN_G;                          // N*128
    float* agg2  = agg1  + (size_t)GN_N * GN_FIN;         // N*64
    float* pool1 = agg2  + (size_t)GN_N * GN_H;           // G*64
    float* pool2 = pool1 + (size_t)GN_G * GN_H;           // G*64
    float* h1    = pool2 + (size_t)GN_G * GN_H;           // N*64
    float* h2    = h1    + (size_t)GN_N * GN_H;           // N*64

    const long long nZero = (long long)GN_N + GN_G + (long long)GN_N * GN_FIN
                          + (long long)GN_N * GN_H + 2LL * GN_G * GN_H;

    const int B = 256;

    // 0. zero accumulators
    ASAP_zero_kernel<<<2048, B, 0, stream>>>(ws, nZero);

    // 1. degrees + batch counts
    ASAP_count_kernel<<<(GN_E + B - 1) / B, B, 0, stream>>>(dst, deg, GN_E);
    ASAP_count_kernel<<<(GN_N + B - 1) / B, B, 0, stream>>>(batch, bc, GN_N);

    // 2. layer-1 edge scatter: agg1[dst] += x[src]  (E*32 float4 chunks)
    {
        long long nWork = (long long)GN_E * (GN_FIN / 4);
        int blocks = (int)((nWork + B - 1) / B);
        ASAP_edge_scatter<<<blocks, B, 0, stream>>>(x, src, dst, agg1, nWork, 5, GN_FIN);
    }

    // 3. h1 = relu((agg1/deg)@W1_rel + b1 + x@W1_root)   [fp32 WMMA]
    {
        int waves = (GN_N / 16) * (GN_H / 16);
        int blocks = (waves * 32 + B - 1) / B;
        ASAP_gemm_dual<<<blocks, B, 0, stream>>>(agg1, W1_rel, GN_FIN,
                                                 x, W1_root, GN_FIN,
                                                 deg, b1_rel, h1, GN_N);
    }

    // 4. pool1 += h1 grouped by batch
    ASAP_pool_scatter<<<(GN_N * GN_H + B - 1) / B, B, 0, stream>>>(h1, batch, pool1, GN_N);

    // 5. layer-2 edge scatter: agg2[dst] += h1[src]  (E*16 float4 chunks)
    {
        long long nWork = (long long)GN_E * (GN_H / 4);
        int blocks = (int)((nWork + B - 1) / B);
        ASAP_edge_scatter<<<blocks, B, 0, stream>>>(h1, src, dst, agg2, nWork, 4, GN_H);
    }

    // 6. h2 = relu((agg2/deg)@W2_rel + b2 + h1@W2_root)  [fp32 WMMA]
    {
        int waves = (GN_N / 16) * (GN_H / 16);
        int blocks = (waves * 32 + B - 1) / B;
        ASAP_gemm_dual<<<blocks, B, 0, stream>>>(agg2, W2_rel, GN_H,
                                                 h1, W2_root, GN_H,
                                                 deg, b2_rel, h2, GN_N);
    }

    // 7. pool2 += h2 grouped by batch, then divide both pools by counts
    ASAP_pool_scatter<<<(GN_N * GN_H + B - 1) / B, B, 0, stream>>>(h2, batch, pool2, GN_N);
    ASAP_pool_finalize<<<(GN_G * GN_H + B - 1) / B, B, 0, stream>>>(pool1, pool2, bc);

    // 8. MLP head + log_softmax
    ASAP_head<<<GN_G, GN_H, 0, stream>>>(pool1, pool2, Wl1, bl1, Wl2, bl2, (float*)d_out);
}